// QuantumConsciousnessLayer_87497073754753
// MI455X (gfx1250) — compile-verified
//
#include <hip/hip_runtime.h>
#include <math.h>

// ---------------------------------------------------------------------------
// QuantumConsciousnessLayer for MI455X (gfx1250, wave32)
//   out = silu(LN( [silu(LN(x@W1+b1)), |qc(x,qw)|^2] @ W2 + b2 ))
// Memory-bound on W1 (256 MB fp32): floor ~11.3 us @ 23.3 TB/s.
// GEMV mapped onto V_WMMA_F32_16X16X4_F32: x chunk in row 0 of the A
// fragment (rows 1..15 zero), W tile streamed into the B fragment via
// coalesced non-temporal b32 loads (W1 > L2, read-once -> TH=NT so x and
// the partials stay L2-resident); D row 0 accumulates 16 outputs in fp32.
// ---------------------------------------------------------------------------

typedef __attribute__((ext_vector_type(2))) float v2f;
typedef __attribute__((ext_vector_type(8))) float v8f;

#define HIDDEN 1024
#define PI_F 3.14159265358979323846f

// ---------------------------------------------------------------------------
// Partial GEMV: partial[kb][j] = sum_{k in chunk kb} X[k] * W[k][j]
// grid.x = N/16/8 tile groups (8 waves/block, one 16-col tile per wave)
// grid.y = K/kchunk, kchunk <= 1024
// ---------------------------------------------------------------------------
__global__ __launch_bounds__(256) void gemv_wmma_partial(
    const float* __restrict__ X, const float* __restrict__ W,
    float* __restrict__ partial, int N, int kchunk)
{
    __shared__ float xs[1024];
    const int tid  = threadIdx.x;
    const int lane = tid & 31;
    const int wave = tid >> 5;
    const int k0   = blockIdx.y * kchunk;

    // stage x chunk in LDS once per block (shared by all 8 waves)
    for (int i = tid; i < kchunk; i += 256) xs[i] = X[k0 + i];
    __syncthreads();

    const int half  = lane >> 4;      // 0: lanes 0-15, 1: lanes 16-31
    const int col   = lane & 15;      // column within the 16-wide tile
    const int ntile = blockIdx.x * 8 + wave;
    const int j0    = ntile * 16;

    // B fragment layout (4x16 f32): VGPR v, lane-half h -> K = 2*h + v, N = col
    const size_t stepN = (size_t)N;
    const float* wp = W + (size_t)(k0 + 2 * half) * stepN + (size_t)(j0 + col);
    const bool  live = (col == 0);    // lanes 0 and 16 carry A row 0

    v8f c = {};
#pragma unroll 8
    for (int kk = 0; kk < kchunk; kk += 4) {
        // Non-temporal: W is read-once streaming data, don't evict L2.
        v2f b;
        b.x = __builtin_nontemporal_load(wp);          // W[k+2h+0][j0+col]
        b.y = __builtin_nontemporal_load(wp + stepN);  // W[k+2h+1][j0+col]
        // A fragment (16x4 f32): VGPR v, lane-half h -> M = lane&15, K = 2*h+v
        v2f a;
        a.x = live ? xs[kk + 2 * half]     : 0.0f;
        a.y = live ? xs[kk + 2 * half + 1] : 0.0f;
        c = __builtin_amdgcn_wmma_f32_16x16x4_f32(
                /*neg_a=*/false, a, /*neg_b=*/false, b,
                /*c_mod=*/(short)0, c, /*reuse_a=*/false, /*reuse_b=*/false);
        wp += 4 * stepN;
    }
    // D row 0 = VGPR 0, lanes 0..15
    if (lane < 16)
        partial[(size_t)blockIdx.y * stepN + (size_t)(j0 + lane)] = c[0];
}

// ---------------------------------------------------------------------------
// Parallel fold of K-split partials: out[g][j] = sum_{i<fold} in[g*fold+i][j]
// grid.x * 256 threads == ngroups * HIDDEN
// ---------------------------------------------------------------------------
__global__ __launch_bounds__(256) void reduce_partials(
    const float* __restrict__ in, float* __restrict__ out, int fold)
{
    const int t = blockIdx.x * 256 + threadIdx.x;
    const int j = t & (HIDDEN - 1);
    const int g = t >> 10;
    const float* p = in + (size_t)g * fold * HIDDEN + j;
    float s = 0.0f;
    for (int i = 0; i < fold; ++i) s += p[(size_t)i * HIDDEN];
    out[(size_t)g * HIDDEN + j] = s;
}

// ---------------------------------------------------------------------------
// Reduce remaining partials, add bias, LayerNorm over 1024, SiLU.
// One block of 1024 threads (thread j owns output j).
// ---------------------------------------------------------------------------
__global__ __launch_bounds__(1024) void finalize_ln_silu(
    const float* __restrict__ partial, int nkb,
    const float* __restrict__ bias, const float* __restrict__ gamma,
    const float* __restrict__ beta, float* __restrict__ out)
{
    __shared__ float red[HIDDEN];
    const int j = threadIdx.x;

    float v = bias[j];
    for (int kb = 0; kb < nkb; ++kb) v += partial[(size_t)kb * HIDDEN + j];

    // mean
    red[j] = v;
    __syncthreads();
    for (int s = HIDDEN / 2; s > 0; s >>= 1) {
        if (j < s) red[j] += red[j + s];
        __syncthreads();
    }
    const float m = red[0] * (1.0f / HIDDEN);
    __syncthreads();

    // variance
    const float d = v - m;
    red[j] = d * d;
    __syncthreads();
    for (int s = HIDDEN / 2; s > 0; s >>= 1) {
        if (j < s) red[j] += red[j + s];
        __syncthreads();
    }
    const float var = red[0] * (1.0f / HIDDEN);

    const float z = gamma[j] * d * __frsqrt_rn(var + 1e-5f) + beta[j];
    out[j] = z / (1.0f + __expf(-z));   // silu
}

// ---------------------------------------------------------------------------
// 10-qubit statevector sim: 1024 amplitudes, one per thread, state in LDS.
// Qubit q <-> bit q of the flat index (matches the reference reshape).
// RX(theta): new_b = cos(t/2)*old_b - i*sin(t/2)*old_{b^1}
// RZ(theta): phase e^{-/+ i theta/2} by bit q.  CZ chain: parity sign.
// ---------------------------------------------------------------------------
__global__ __launch_bounds__(1024) void quantum_kernel(
    const float* __restrict__ x, const float* __restrict__ qw,
    float* __restrict__ probs)
{
    __shared__ float sre[1024];
    __shared__ float sim_[1024];
    const int t = threadIdx.x;

    float re = (t == 0) ? 1.0f : 0.0f;
    float im = 0.0f;

    // encoding layer: RX(a) then RZ(a/2), a = sigmoid(x[q])*pi
    for (int q = 0; q < 10; ++q) {
        const float a = PI_F / (1.0f + __expf(-x[q]));
        float sh, ch;
        __sincosf(0.5f * a, &sh, &ch);

        sre[t] = re; sim_[t] = im;
        __syncthreads();
        const int p = t ^ (1 << q);
        const float pre = sre[p], pim = sim_[p];
        __syncthreads();
        // (-i*s)*(pre + i*pim) = s*pim - i*s*pre
        re = ch * re + sh * pim;
        im = ch * im - sh * pre;

        // RZ(a/2): angle/2 = a/4; bit0 -> e^{-i a/4}, bit1 -> e^{+i a/4}
        float s2, c2;
        __sincosf(0.25f * a, &s2, &c2);
        const float sgn = ((t >> q) & 1) ? 1.0f : -1.0f;
        const float nr = re * c2 - im * (sgn * s2);
        const float ni = im * c2 + re * (sgn * s2);
        re = nr; im = ni;
    }

    // 3 variational layers: RX(qw*pi) per qubit, then CZ ladder signs
    int pix = 0;
    for (int layer = 0; layer < 3; ++layer) {
        for (int q = 0; q < 10; ++q) {
            const float a = qw[pix++] * PI_F;
            float sh, ch;
            __sincosf(0.5f * a, &sh, &ch);
            sre[t] = re; sim_[t] = im;
            __syncthreads();
            const int p = t ^ (1 << q);
            const float pre = sre[p], pim = sim_[p];
            __syncthreads();
            re = ch * re + sh * pim;
            im = ch * im - sh * pre;
        }
        float s = 1.0f;
        for (int c = 0; c < 9; ++c)
            if (((t >> c) & 1) & ((t >> (c + 1)) & 1)) s = -s;
        re *= s; im *= s;
    }

    probs[t] = re * re + im * im;
}

// ---------------------------------------------------------------------------
// Host-side launch (graph-capture safe: only kernel launches on `stream`)
// ---------------------------------------------------------------------------
extern "C" void kernel_launch(void* const* d_in, const int* in_sizes, int n_in,
                              void* d_out, int out_size, void* d_ws, size_t ws_size,
                              hipStream_t stream)
{
    (void)in_sizes; (void)n_in; (void)out_size; (void)ws_size;

    const float* x   = (const float*)d_in[0];
    const float* W1  = (const float*)d_in[1];
    const float* b1  = (const float*)d_in[2];
    const float* b1g = (const float*)d_in[3];
    const float* be1 = (const float*)d_in[4];
    const float* qw  = (const float*)d_in[5];
    const float* W2  = (const float*)d_in[6];
    const float* b2  = (const float*)d_in[7];
    const float* b2g = (const float*)d_in[8];
    const float* be2 = (const float*)d_in[9];
    float* out = (float*)d_out;

    // workspace layout (floats)
    float* ws        = (float*)d_ws;
    const int NKB1   = 128;                 // 65536 / 512 K-split blocks
    const int NGRP   = 8;                   // fold 128 -> 8 partials
    const int NKB2   = 4;                   // 2048 / 512
    float* partial1  = ws;                         // NKB1*1024
    float* stage2    = partial1 + NKB1 * HIDDEN;   // NGRP*1024
    float* combined  = stage2 + NGRP * HIDDEN;     // 2048 (classical|quantum)
    float* classical = combined;
    float* quantum   = combined + HIDDEN;
    float* partial2  = combined + 2 * HIDDEN;      // NKB2*1024

    // 1) classical = silu(LN(x @ W1 + b1))
    gemv_wmma_partial<<<dim3(HIDDEN / 16 / 8, NKB1), dim3(256), 0, stream>>>(
        x, W1, partial1, HIDDEN, 512);
    reduce_partials<<<dim3(NGRP * HIDDEN / 256), dim3(256), 0, stream>>>(
        partial1, stage2, NKB1 / NGRP);
    finalize_ln_silu<<<dim3(1), dim3(HIDDEN), 0, stream>>>(
        stage2, NGRP, b1, b1g, be1, classical);

    // 2) quantum features = |qc(x, qw)|^2
    quantum_kernel<<<dim3(1), dim3(1024), 0, stream>>>(x, qw, quantum);

    // 3) out = silu(LN(combined @ W2 + b2))
    gemv_wmma_partial<<<dim3(HIDDEN / 16 / 8, NKB2), dim3(256), 0, stream>>>(
        combined, W2, partial2, HIDDEN, 512);
    finalize_ln_silu<<<dim3(1), dim3(HIDDEN), 0, stream>>>(
        partial2, NKB2, b2, b2g, be2, out);
}